// TableEmbed3D_22840636080898
// MI455X (gfx1250) — compile-verified
//
#include <hip/hip_runtime.h>

typedef float v4f __attribute__((ext_vector_type(4)));

#define TS   32            // spatial size per dim
#define EMB  64            // channels
#define VOX  (TS * TS * TS)  // 32768 voxels

#if __has_builtin(__builtin_amdgcn_global_load_async_to_lds_b32)
#define HAVE_ASYNC_LDS 1
#endif

// ---------------------------------------------------------------------------
// Axis setup, mirroring the reference exactly:
//   pix = ((clip(c,-1,1) + 1) * 32 - 1) * 0.5 ;  p0 = floor(pix); f = pix-p0
// Zeros-padding folded into per-axis weights (w=0 when the corner is OOB),
// index clamped to [0,31] (matches reference idxc + inb masking).
// ---------------------------------------------------------------------------
__device__ __forceinline__ void axis_setup(float c, float& w0, float& w1,
                                           int& i0, int& i1) {
  c = fminf(fmaxf(c, -1.0f), 1.0f);
  float pix = ((c + 1.0f) * 32.0f - 1.0f) * 0.5f;   // (c+1)*32 exact scale
  float pf  = floorf(pix);
  float f   = pix - pf;
  int p0 = (int)pf;          // in [-1, 31]
  int p1 = p0 + 1;           // in [ 0, 32]
  w0 = (1.0f - f) * (p0 >= 0 ? 1.0f : 0.0f);
  w1 = f * (p1 <= TS - 1 ? 1.0f : 0.0f);
  i0 = p0 < 0 ? 0 : p0;
  i1 = p1 > TS - 1 ? TS - 1 : p1;
}

// ---------------------------------------------------------------------------
// Transpose table [E, VOX] -> [VOX, E] so a corner's 64 channels are one
// contiguous 256B row. LDS-tiled for coalescing on both sides.
// Grid: VOX/64 blocks of 256 threads; each block handles 64 voxels.
// ---------------------------------------------------------------------------
__global__ __launch_bounds__(256) void table_transpose_kernel(
    const float* __restrict__ tin, float* __restrict__ tout) {
  __shared__ float tile[EMB * 65];  // pad 65 to dodge bank conflicts
  const int tid   = threadIdx.x;
  const int vbase = blockIdx.x * 64;
#pragma unroll
  for (int k = 0; k < 16; ++k) {
    int e  = k * 4 + (tid >> 6);
    int vl = tid & 63;
    tile[e * 65 + vl] = tin[(size_t)e * VOX + vbase + vl];  // coalesced over v
  }
  __syncthreads();
#pragma unroll
  for (int k = 0; k < 16; ++k) {
    int vl = k * 4 + (tid >> 6);
    int e  = tid & 63;
    tout[(size_t)(vbase + vl) * EMB + e] = tile[e * 65 + vl];  // coalesced over e
  }
}

// ---------------------------------------------------------------------------
// Main sampler: 8 lanes per point (lane owns channels [l*4,l*4+4) and
// [32+l*4, 32+l*4+4)), 4 points per wave32, 32 points per 256-thread block.
// Table rows are contiguous [v*64 .. v*64+63]; each (z,y) pair needs 2 rows.
// Output written with non-temporal stores (256MB stream; keep table in L2).
// ---------------------------------------------------------------------------
__global__ __launch_bounds__(256) void grid_sample3d_kernel(
    const float* __restrict__ x, const float* __restrict__ tab,
    float* __restrict__ out, int n) {
  const int tid    = threadIdx.x;
  const int pblock = blockIdx.x * 32;
  const int g      = tid >> 3;   // point-in-block 0..31
  const int lane   = tid & 7;
  const int p      = pblock + g;

  // ---- stage this block's 32x3 coords into LDS (one coalesced 384B read) ----
  __shared__ float scoord[96];
  int nflt = n - pblock;
  nflt = (nflt > 32 ? 32 : nflt) * 3;
#ifdef HAVE_ASYNC_LDS
  if (tid < nflt) {
    __builtin_amdgcn_global_load_async_to_lds_b32(
        (__attribute__((address_space(1))) int*)(x + (size_t)pblock * 3 + tid),
        (__attribute__((address_space(3))) int*)&scoord[tid],
        /*offset=*/0, /*cpol=*/0);
  }
#if __has_builtin(__builtin_amdgcn_s_wait_asynccnt)
  __builtin_amdgcn_s_wait_asynccnt(0);
#else
  asm volatile("s_wait_asynccnt 0" ::: "memory");
#endif
#else
  if (tid < nflt) scoord[tid] = x[(size_t)pblock * 3 + tid];
#endif
  __syncthreads();

  if (p >= n) return;

  const float cx = scoord[g * 3 + 0];
  const float cy = scoord[g * 3 + 1];
  const float cz = scoord[g * 3 + 2];

  float wx0, wx1, wy0, wy1, wz0, wz1;
  int ix0, ix1, iy0, iy1, iz0, iz1;
  axis_setup(cx, wx0, wx1, ix0, ix1);
  axis_setup(cy, wy0, wy1, iy0, iy1);
  axis_setup(cz, wz0, wz1, iz0, iz1);

  const int l4  = lane * 4;
  const int rx0 = ix0 * EMB + l4;
  const int rx1 = ix1 * EMB + l4;

  v4f acc0 = {0.f, 0.f, 0.f, 0.f};
  v4f acc1 = {0.f, 0.f, 0.f, 0.f};

#pragma unroll
  for (int dz = 0; dz < 2; ++dz) {
    const float wz = dz ? wz1 : wz0;
    const int   zo = (dz ? iz1 : iz0) * (TS * TS * EMB);
#pragma unroll
    for (int dy = 0; dy < 2; ++dy) {
      const float wzy  = wz * (dy ? wy1 : wy0);
      const int   base = zo + (dy ? iy1 : iy0) * (TS * EMB);
      const float a0   = wzy * wx0;
      const float a1   = wzy * wx1;
      const float* r0  = tab + base + rx0;
      const float* r1  = tab + base + rx1;
      v4f A0 = *(const v4f*)(r0);        // global_load_b128, 8 lanes = 128B
      v4f A1 = *(const v4f*)(r0 + 32);
      v4f B0 = *(const v4f*)(r1);
      v4f B1 = *(const v4f*)(r1 + 32);
      acc0 += A0 * a0 + B0 * a1;
      acc1 += A1 * a0 + B1 * a1;
    }
  }

  float* o = out + (size_t)p * EMB + l4;
  __builtin_nontemporal_store(acc0, (v4f*)o);          // streaming store
  __builtin_nontemporal_store(acc1, (v4f*)(o + 32));
}

// ---------------------------------------------------------------------------
// Fallback (workspace too small for the 8MB transposed table): sample the
// original [E, VOX] layout directly. One thread per (point, channel).
// ---------------------------------------------------------------------------
__global__ __launch_bounds__(256) void grid_sample3d_fallback_kernel(
    const float* __restrict__ x, const float* __restrict__ tab,
    float* __restrict__ out, int n) {
  const long long t = (long long)blockIdx.x * blockDim.x + threadIdx.x;
  if (t >= (long long)n * EMB) return;
  const int p = (int)(t >> 6);
  const int e = (int)(t & 63);

  const float cx = x[(size_t)p * 3 + 0];
  const float cy = x[(size_t)p * 3 + 1];
  const float cz = x[(size_t)p * 3 + 2];

  float wx0, wx1, wy0, wy1, wz0, wz1;
  int ix0, ix1, iy0, iy1, iz0, iz1;
  axis_setup(cx, wx0, wx1, ix0, ix1);
  axis_setup(cy, wy0, wy1, iy0, iy1);
  axis_setup(cz, wz0, wz1, iz0, iz1);

  const float* te = tab + (size_t)e * VOX;
  float acc = 0.f;
#pragma unroll
  for (int dz = 0; dz < 2; ++dz) {
    const float wz = dz ? wz1 : wz0;
    const int   zb = (dz ? iz1 : iz0) * TS;
#pragma unroll
    for (int dy = 0; dy < 2; ++dy) {
      const float wzy = wz * (dy ? wy1 : wy0);
      const int   yb  = (zb + (dy ? iy1 : iy0)) * TS;
      acc += te[yb + ix0] * (wzy * wx0);
      acc += te[yb + ix1] * (wzy * wx1);
    }
  }
  out[t] = acc;
}

// ---------------------------------------------------------------------------
extern "C" void kernel_launch(void* const* d_in, const int* in_sizes, int n_in,
                              void* d_out, int out_size, void* d_ws, size_t ws_size,
                              hipStream_t stream) {
  const float* x     = (const float*)d_in[0];
  const float* table = (const float*)d_in[1];
  float*       out   = (float*)d_out;
  const int    n     = in_sizes[0] / 3;

  const size_t need = (size_t)VOX * EMB * sizeof(float);  // 8 MB
  if (ws_size >= need) {
    float* tws = (float*)d_ws;
    table_transpose_kernel<<<VOX / 64, 256, 0, stream>>>(table, tws);
    grid_sample3d_kernel<<<(n + 31) / 32, 256, 0, stream>>>(x, tws, out, n);
  } else {
    const long long total = (long long)n * EMB;
    grid_sample3d_fallback_kernel<<<(unsigned)((total + 255) / 256), 256, 0, stream>>>(
        x, table, out, n);
  }
}